// CIN_47021301957241
// MI455X (gfx1250) — compile-verified
//
#include <hip/hip_runtime.h>

typedef __attribute__((ext_vector_type(2))) float v2f;
typedef __attribute__((ext_vector_type(8))) float v8f;

#define NB   1024   // batch
#define MF   40     // fields
#define DD   64     // embedding dim
#define HH   128    // HK
#define KP   132    // padded row stride for transposed x / xk in LDS ([d][j])
#define WTP  68     // padded row stride for transposed W tile in LDS ([h][j])
#define JT   64     // max W tile rows (j per tile)

// LDS partition (floats):
//   sX   : 64*132       = 8448   (x transposed:  sX[d*KP + i])
//   sK   : 64*132       = 8448   (xk transposed: sK[d*KP + j])
//   sWt  : 2 * 128*68   = 17408  (double-buffered W tile, TRANSPOSED [h][j])
//   sOut : 128
// total = 34432 floats = 137728 bytes

template<int J, int JTC, int TPI, bool L0, bool HASY>
__global__ __launch_bounds__(256) void cin_layer(
    const float* __restrict__ x,     // [B][40][64]
    const float* __restrict__ xk,    // [B][64][J] d-major (ignored when L0)
    const float* __restrict__ W,     // [40][J][128]
    float* __restrict__ yout,        // [B][64][128] d-major (ignored when !HASY)
    float* __restrict__ out,         // [B][384]
    int outOfs)
{
  extern __shared__ float smem[];
  float* sX   = smem;
  float* sK   = smem + DD * KP;
  float* sWt  = smem + 2 * DD * KP;
  float* sOut = smem + 2 * DD * KP + 2 * HH * WTP;

  const int b    = blockIdx.x;
  const int tid  = threadIdx.x;
  const int lane = tid & 31;
  const int wid  = tid >> 5;
  const int m    = lane & 15;   // tile column / A-row select
  const int hlf  = lane >> 4;   // which K-pair half this lane provides
  const int dBase  = (wid & 3) * 16;     // wave's 16-row d strip
  const int dLane  = dBase + m;          // this lane's A-matrix row (d)
  const int hBase0 = (wid >> 2) * 64;    // wave covers h tiles hBase0 + 16*t

  // ---- stage x transposed into sX: sX[d*KP + i] ----
  const float* xb = x + (size_t)b * MF * DD;
  for (int e = tid; e < MF * DD; e += 256) {
    int i = e >> 6, d = e & 63;
    sX[d * KP + i] = xb[e];
  }
  // ---- stage xk transposed into sK: sK[d*KP + j] ----
  if constexpr (!L0) {
    const float4* src = (const float4*)(xk + (size_t)b * DD * HH);
    #pragma unroll
    for (int q = 0; q < DD * HH / 4 / 256; ++q) {
      int e = tid + q * 256;
      int d = e >> 5, j4 = e & 31;
      *(float4*)&sK[d * KP + j4 * 4] = src[e];
    }
  } else {
    for (int e = tid; e < MF * DD; e += 256) {
      int j = e >> 6, d = e & 63;
      sK[d * KP + j] = xb[e];
    }
  }
  if (tid < HH) sOut[tid] = 0.0f;

  constexpr int NTILES = MF * TPI;
  constexpr int NG     = JTC * HH / 4 / 256;  // 4-float column groups per thread

  v8f acc[4];
  acc[0] = v8f{}; acc[1] = v8f{}; acc[2] = v8f{}; acc[3] = v8f{};

  // thread's (h, j4) slot inside a tile, fixed across tiles
  // group g = tid + q*256 : h = g & 127, j4 = (g >> 7) * 4
  float pf[NG * 4];

  // prefetch tile 0 (i=0, jt0=0): read 4 along j so the LDS store is one b128
  {
    #pragma unroll
    for (int q = 0; q < NG; ++q) {
      int g = tid + q * 256;
      int h = g & 127, j4 = (g >> 7) * 4;
      const float* src = W + (size_t)j4 * HH + h;
      pf[4*q+0] = src[0];
      pf[4*q+1] = src[HH];
      pf[4*q+2] = src[2*HH];
      pf[4*q+3] = src[3*HH];
    }
  }

  __syncthreads();  // sX/sK staged

  int buf = 0;
  for (int tile = 0; tile < NTILES; ++tile) {
    const int i   = (TPI == 2) ? (tile >> 1) : tile;
    const int jt0 = (TPI == 2) ? ((tile & 1) * JTC) : 0;
    float* sWc = sWt + buf * (HH * WTP);

    // commit prefetched tile to LDS, transposed: sWc[h*WTP + j]
    #pragma unroll
    for (int q = 0; q < NG; ++q) {
      int g = tid + q * 256;
      int h = g & 127, j4 = (g >> 7) * 4;
      *(float4*)&sWc[h * WTP + j4] = float4{pf[4*q+0], pf[4*q+1], pf[4*q+2], pf[4*q+3]};
    }
    // issue global loads for the next tile (overlaps with compute below)
    if (tile + 1 < NTILES) {
      const int i1   = (TPI == 2) ? ((tile + 1) >> 1) : (tile + 1);
      const int jt1  = (TPI == 2) ? (((tile + 1) & 1) * JTC) : 0;
      const float* g0 = W + ((size_t)i1 * J + jt1) * HH;
      #pragma unroll
      for (int q = 0; q < NG; ++q) {
        int g = tid + q * 256;
        int h = g & 127, j4 = (g >> 7) * 4;
        const float* src = g0 + (size_t)j4 * HH + h;
        pf[4*q+0] = src[0];
        pf[4*q+1] = src[HH];
        pf[4*q+2] = src[2*HH];
        pf[4*q+3] = src[3*HH];
      }
    }
    // warm L2 for the tile after next (uniform address, one line is enough to
    // kick the stride prefetcher; global_prefetch_b8)
    if (tile + 2 < NTILES) {
      const int i2  = (TPI == 2) ? ((tile + 2) >> 1) : (tile + 2);
      const int jt2 = (TPI == 2) ? (((tile + 2) & 1) * JTC) : 0;
      __builtin_prefetch(W + ((size_t)i2 * J + jt2) * HH, 0, 1);
    }
    __syncthreads();

    // ---- WMMA compute over this tile's K range ----
    const float  x0v   = sX[dLane * KP + i];               // hoisted per tile
    const float* kRow  = &sK[dLane * KP + jt0 + 2 * hlf];  // xk[jA..jA+1][d]
    const float* wBase = &sWc[(hBase0 + m) * WTP + 2 * hlf];

    #pragma unroll 4
    for (int j4 = 0; j4 < JTC; j4 += 4) {
      // A fragment: A[d][k] = x0[i][d] * xk[j_k][d]   (one ds_load_b64)
      v2f kv = *(const v2f*)(kRow + j4);
      v2f a;
      a.x = x0v * kv.x;
      a.y = x0v * kv.y;
      #pragma unroll
      for (int t = 0; t < 4; ++t) {
        // B fragment: one ds_load_b64 from transposed W tile
        v2f bf = *(const v2f*)(wBase + t * 16 * WTP + j4);
        acc[t] = __builtin_amdgcn_wmma_f32_16x16x4_f32(
            false, a, false, bf, (short)0, acc[t], false, false);
      }
    }
    buf ^= 1;
  }

  // ---- epilogue: write next-layer xk (d-major) ----
  if constexpr (HASY) {
    float* yb = yout + (size_t)b * DD * HH;
    #pragma unroll
    for (int t = 0; t < 4; ++t) {
      int hb = hBase0 + 16 * t;
      #pragma unroll
      for (int r = 0; r < 8; ++r) {
        int d = dBase + hlf * 8 + r;   // C layout: VGPR r -> row r (+8 upper half)
        yb[d * HH + hb + m] = acc[t][r];
      }
    }
  }

  // ---- sum-pool over d into out[b, outOfs + h] ----
  #pragma unroll
  for (int t = 0; t < 4; ++t) {
    float p = acc[t][0] + acc[t][1] + acc[t][2] + acc[t][3]
            + acc[t][4] + acc[t][5] + acc[t][6] + acc[t][7];
    atomicAdd(&sOut[hBase0 + 16 * t + m], p);   // ds_add_f32
  }
  __syncthreads();
  if (tid < HH) out[(size_t)b * 384 + outOfs + tid] = sOut[tid];
}

extern "C" void kernel_launch(void* const* d_in, const int* in_sizes, int n_in,
                              void* d_out, int out_size, void* d_ws, size_t ws_size,
                              hipStream_t stream) {
  const float* x  = (const float*)d_in[0];  // [1024][40][64]
  const float* W0 = (const float*)d_in[1];  // [40][40][128]
  const float* W1 = (const float*)d_in[2];  // [40][128][128]
  const float* W2 = (const float*)d_in[3];  // [40][128][128]
  float* out = (float*)d_out;               // [1024][384]

  float* y0 = (float*)d_ws;                       // [1024][64][128]
  float* y1 = y0 + (size_t)NB * DD * HH;          // [1024][64][128]

  const size_t shmem = (size_t)(2 * DD * KP + 2 * HH * WTP + 128) * sizeof(float);
  dim3 grid(NB), block(256);
  cin_layer<MF, MF, 1, true,  true ><<<grid, block, shmem, stream>>>(x, x,  W0, y0, out, 0);
  cin_layer<HH, JT, 2, false, true ><<<grid, block, shmem, stream>>>(x, y0, W1, y1, out, 128);
  cin_layer<HH, JT, 2, false, false><<<grid, block, shmem, stream>>>(x, y1, W2, nullptr, out, 256);
}